// MatchingLayer_13511967113276
// MI455X (gfx1250) — compile-verified
//
#include <hip/hip_runtime.h>

typedef __attribute__((ext_vector_type(2))) float v2f;
typedef __attribute__((ext_vector_type(8))) float v8f;

#define EPSV 1e-6f
#define DD   100   // feature dim per direction
#define PP   10    // perspectives
#define SEQ  100   // sequence length (both sides)
#define BB   256   // batch
#define MT   112   // 7*16 padded tile extent
#define LD   104   // LDS row stride for S1/S2/ATT/WSQ (floats)
#define CLD  114   // LDS row stride for cos buffer (even -> 8B-aligned v2f loads)

// LDS layout (float offsets)
#define OFF_S1   0
#define OFF_S2   (OFF_S1  + MT*LD)        // 11648
#define OFF_COS  (OFF_S2  + MT*LD)        // 23296
#define OFF_ATT  (OFF_COS + MT*CLD)       // 36064
#define OFF_WSQ  (OFF_ATT + SEQ*LD)       // 46464  (4 weight sets * PP * LD)
#define OFF_N1   (OFF_WSQ + 4*PP*LD)      // 50624
#define OFF_N2   (OFF_N1  + MT)
#define OFF_N1W  (OFF_N2  + MT)           // 4*SEQ*PP
#define OFF_N2W  (OFF_N1W + 4*SEQ*PP)     // SEQ*PP (maxpool s2 norms)
#define OFF_RS   (OFF_N2W + SEQ*PP)       // SEQ row sums
#define OFF_IDX  (OFF_RS  + SEQ)          // SEQ argmax (int)
#define SMEM_FLOATS (OFF_IDX + SEQ)

__device__ __forceinline__ v8f wmma_f32(v2f a, v2f b, v8f c) {
  // D = A(16x4) * B(4x16) + C, fp32, wave32
  return __builtin_amdgcn_wmma_f32_16x16x4_f32(false, a, false, b, (short)0, c,
                                               false, false);
}

__global__ __launch_bounds__(256)
void matching_layer_kernel(const float* __restrict__ s1g,
                           const float* __restrict__ s2g,
                           const float* __restrict__ w1, const float* __restrict__ w2,
                           const float* __restrict__ w3, const float* __restrict__ w4,
                           const float* __restrict__ w5, const float* __restrict__ w6,
                           const float* __restrict__ w7, const float* __restrict__ w8,
                           float* __restrict__ out)
{
  extern __shared__ float sm[];
  float* S1  = sm + OFF_S1;
  float* S2  = sm + OFF_S2;
  float* COS = sm + OFF_COS;
  float* ATT = sm + OFF_ATT;
  float* WQ  = sm + OFF_WSQ;   // WQ[(v*PP+p)*LD + d] = w^2 ; v: 0=full 1=maxp 2=mean 3=last
  float* N1  = sm + OFF_N1;
  float* N2  = sm + OFF_N2;
  float* N1W = sm + OFF_N1W;   // [v*1000 + i*10 + p]
  float* N2W = sm + OFF_N2W;   // maxpool only: [j*10 + p]
  float* RS  = sm + OFF_RS;
  int*   IDX = (int*)(sm + OFF_IDX);

  const int tid  = threadIdx.x;
  const int wv   = tid >> 5;          // wave id (8 waves)
  const int lane = tid & 31;
  const int m16  = lane & 15;
  const int kh   = lane >> 4;         // 0/1 half-wave selector (K pairs)

  const int blk = blockIdx.x;         // 512 blocks
  const int b   = blk & 255;
  const int dir = blk >> 8;           // 0 = forward half, 1 = backward half
  const int doff = dir * DD;

  const float* wsel[4];
  if (dir == 0) { wsel[0] = w1; wsel[1] = w3; wsel[2] = w5; wsel[3] = w7; }
  else          { wsel[0] = w2; wsel[1] = w4; wsel[2] = w6; wsel[3] = w8; }

  // ---------------- Phase 0: load S1, S2 (zero-padded) + squared weights ----
  for (int t = tid; t < MT * LD; t += 256) {
    int r = t / LD, c = t % LD;
    float v1 = 0.f, v2 = 0.f;
    if (r < SEQ && c < DD) {
      size_t g = ((size_t)r * BB + b) * (2 * DD) + doff + c;
      v1 = s1g[g];
      v2 = s2g[g];
    }
    S1[t] = v1;
    S2[t] = v2;
  }
  for (int t = tid; t < 4 * PP * LD; t += 256) {
    int v = t / (PP * LD), rem = t % (PP * LD);
    int p = rem / LD, c = rem % LD;
    float wvv = (c < DD) ? wsel[v][p * DD + c] : 0.f;
    WQ[t] = wvv * wvv;
  }
  __syncthreads();

  // ---------------- Phase 1: norms + "full" outputs -------------------------
  {
    const int T_N = 2 * MT, T_N1W = 4 * SEQ * PP, T_N2W = SEQ * PP, T_FULL = SEQ * PP;
    const int T_TOT = T_N + T_N1W + T_N2W + T_FULL;
    for (int t = tid; t < T_TOT; t += 256) {
      if (t < T_N) {
        int which = t / MT, r = t % MT;
        const float* Sx = which ? S2 : S1;
        float s = 0.f;
        for (int d = 0; d < DD; ++d) { float x = Sx[r * LD + d]; s += x * x; }
        (which ? N2 : N1)[r] = sqrtf(fmaxf(s, EPSV));
      } else if (t < T_N + T_N1W) {
        int u = t - T_N;
        int v = u / (SEQ * PP), i = (u % (SEQ * PP)) / PP, p = u % PP;
        const float* wq = &WQ[(v * PP + p) * LD];
        const float* r  = &S1[i * LD];
        float s = 0.f;
        for (int d = 0; d < DD; ++d) s += r[d] * r[d] * wq[d];
        N1W[v * (SEQ * PP) + i * PP + p] = sqrtf(fmaxf(s, EPSV));
      } else if (t < T_N + T_N1W + T_N2W) {
        int u = t - T_N - T_N1W;
        int j = u / PP, p = u % PP;
        const float* wq = &WQ[(1 * PP + p) * LD];
        const float* r  = &S2[j * LD];
        float s = 0.f;
        for (int d = 0; d < DD; ++d) s += r[d] * r[d] * wq[d];
        N2W[j * PP + p] = sqrtf(fmaxf(s, EPSV));
      } else {
        int u = t - T_N - T_N1W - T_N2W;
        int i = u / PP, p = u % PP;
        const float* wq = &WQ[(0 * PP + p) * LD];
        const float* r  = &S1[i * LD];
        const float* l  = &S2[(SEQ - 1) * LD];   // s2[-1]
        float num = 0.f, na = 0.f, nb = 0.f;
        for (int d = 0; d < DD; ++d) {
          float wd = wq[d];
          num += r[d] * l[d] * wd;
          na  += r[d] * r[d] * wd;
          nb  += l[d] * l[d] * wd;
        }
        out[((size_t)i * BB + b) * 80 + dir * 10 + p] =
            num / (sqrtf(fmaxf(na, EPSV)) * sqrtf(fmaxf(nb, EPSV)));
      }
    }
  }
  __syncthreads();

  // ---------------- Phase 2: G = S1 * S2^T  (WMMA fp32), cos epilogue -------
  for (int t = wv; t < 49; t += 8) {
    int mt = t / 7, nt = t % 7;
    v8f acc = {0.f, 0.f, 0.f, 0.f, 0.f, 0.f, 0.f, 0.f};
    const float* arow = &S1[(mt * 16 + m16) * LD];
    const float* brow = &S2[(nt * 16 + m16) * LD];
    for (int k0 = 0; k0 < DD; k0 += 4) {
      v2f a = *(const v2f*)(arow + k0 + 2 * kh);
      v2f bf = *(const v2f*)(brow + k0 + 2 * kh);
      acc = wmma_f32(a, bf, acc);
    }
    int col = nt * 16 + m16;
    float n2c = N2[col];
    for (int r = 0; r < 8; ++r) {
      int row = mt * 16 + r + 8 * kh;
      COS[row * CLD + col] = acc[r] / (N1[row] * n2c);
    }
  }
  __syncthreads();

  // ---------------- Phase 2b: row sums + argmax over j ----------------------
  if (tid < SEQ) {
    int i = tid;
    float s = 0.f, best = -3.4e38f;
    int bj = 0;
    for (int j = 0; j < SEQ; ++j) {
      float c = COS[i * CLD + j];
      s += c;
      if (c > best) { best = c; bj = j; }
    }
    RS[i] = s;
    IDX[i] = bj;
  }
  __syncthreads();

  // ---------------- Phase 3: att = cos * S2 (WMMA fp32) ---------------------
  for (int t = wv; t < 49; t += 8) {
    int mt = t / 7, nt = t % 7;
    v8f acc = {0.f, 0.f, 0.f, 0.f, 0.f, 0.f, 0.f, 0.f};
    const float* arow = &COS[(mt * 16 + m16) * CLD];
    int coln = nt * 16 + m16;
    for (int k0 = 0; k0 < MT; k0 += 4) {        // padded cos cols are exactly 0
      v2f a = *(const v2f*)(arow + k0 + 2 * kh);
      v2f bf;
      bf.x = S2[(k0 + 2 * kh) * LD + coln];
      bf.y = S2[(k0 + 2 * kh + 1) * LD + coln];
      acc = wmma_f32(a, bf, acc);
    }
    if (coln < DD) {
      for (int r = 0; r < 8; ++r) {
        int row = mt * 16 + r + 8 * kh;
        if (row < SEQ) ATT[row * LD + coln] = acc[r];
      }
    }
  }
  __syncthreads();
  for (int t = tid; t < SEQ * DD; t += 256) {
    int i = t / DD, d = t % DD;
    ATT[i * LD + d] = ATT[i * LD + d] / (RS[i] + EPSV);
  }
  __syncthreads();

  // ---------------- Phase 4: mean-att + max-att/mean-att(w8) outputs --------
  for (int t = tid; t < 2 * SEQ * PP; t += 256) {
    int kind = t / (SEQ * PP);
    int rem = t % (SEQ * PP);
    int i = rem / PP, p = rem % PP;
    const float* wq;
    const float* vec;
    int vsel;
    if (kind == 0) { wq = &WQ[(2 * PP + p) * LD]; vec = &ATT[i * LD]; vsel = 2; }
    else {
      wq = &WQ[(3 * PP + p) * LD];
      vec = (dir == 0) ? &S2[IDX[i] * LD] : &ATT[i * LD];   // f: gather argmax; b: mean-att(w8)
      vsel = 3;
    }
    const float* s1r = &S1[i * LD];
    float num = 0.f, na = 0.f;
    for (int d = 0; d < DD; ++d) {
      float wd = wq[d], av = vec[d], sv = s1r[d];
      num += sv * av * wd;
      na  += av * av * wd;
    }
    float ns = N1W[vsel * (SEQ * PP) + i * PP + p];
    out[((size_t)i * BB + b) * 80 + (kind == 0 ? 40 : 60) + dir * 10 + p] =
        num / (ns * sqrtf(fmaxf(na, EPSV)));
  }
  __syncthreads();

  // ---------------- Phase 5: maxpool — 10 weighted WMMA GEMMs + row max -----
  for (int p = 0; p < PP; ++p) {
    const float* wq = &WQ[(1 * PP + p) * LD];
    for (int t = wv; t < 49; t += 8) {
      int mt = t / 7, nt = t % 7;
      v8f acc = {0.f, 0.f, 0.f, 0.f, 0.f, 0.f, 0.f, 0.f};
      const float* arow = &S1[(mt * 16 + m16) * LD];
      const float* brow = &S2[(nt * 16 + m16) * LD];
      for (int k0 = 0; k0 < DD; k0 += 4) {
        int ka = k0 + 2 * kh;
        v2f a = *(const v2f*)(arow + ka);
        v2f wqa = *(const v2f*)(wq + ka);
        a *= wqa;                                   // fold w^2 into A
        v2f bf = *(const v2f*)(brow + ka);
        acc = wmma_f32(a, bf, acc);
      }
      int col = nt * 16 + m16;
      float dn = (col < SEQ) ? N2W[col * PP + p] : 1.0f;
      for (int r = 0; r < 8; ++r) {
        int row = mt * 16 + r + 8 * kh;
        float ns = (row < SEQ) ? N1W[1 * (SEQ * PP) + row * PP + p] : 1.0f;
        COS[row * CLD + col] = acc[r] / (ns * dn);
      }
    }
    __syncthreads();
    if (tid < SEQ) {
      int i = tid;
      float best = -3.4e38f;
      for (int j = 0; j < SEQ; ++j) best = fmaxf(best, COS[i * CLD + j]);
      out[((size_t)i * BB + b) * 80 + 20 + dir * 10 + p] = best;
    }
    __syncthreads();
  }
}

extern "C" void kernel_launch(void* const* d_in, const int* in_sizes, int n_in,
                              void* d_out, int out_size, void* d_ws, size_t ws_size,
                              hipStream_t stream) {
  const float* s1 = (const float*)d_in[0];
  const float* s2 = (const float*)d_in[1];
  const float* w[8];
  for (int i = 0; i < 8; ++i) w[i] = (const float*)d_in[2 + i];
  float* out = (float*)d_out;

  size_t smem = (size_t)SMEM_FLOATS * sizeof(float);   // ~219 KB (<= 320 KB/WGP)
  (void)hipFuncSetAttribute((const void*)matching_layer_kernel,
                            hipFuncAttributeMaxDynamicSharedMemorySize, (int)smem);

  matching_layer_kernel<<<dim3(512), dim3(256), smem, stream>>>(
      s1, s2, w[0], w[1], w[2], w[3], w[4], w[5], w[6], w[7], out);
}